// Damp_20375324852241
// MI455X (gfx1250) — compile-verified
//
#include <hip/hip_runtime.h>
#include <hip/hip_bf16.h>

// ---- problem constants (from reference) ----
#define BB   3
#define NN   6
#define DD   48
#define DH   28
#define DW   60
#define CC   64
#define XD   200
#define YD   200
#define PER_BN   (DD * DH * DW)                  // 80640, divisible by 16
#define NPTS     ((long long)BB * NN * PER_BN)   // 1,451,520
#define NCHUNK   ((int)(NPTS / 16))              // 90,720
#define ACC_ELEMS ((size_t)BB * XD * YD * CC)    // 7,680,000 floats

typedef float v2f __attribute__((ext_vector_type(2)));
typedef float v8f __attribute__((ext_vector_type(8)));

__device__ __forceinline__ float pick3(int r, float x0, float x1, float x2) {
    return (r == 0) ? x0 : (r == 1) ? x1 : (r == 2) ? x2 : 0.0f;
}

// One wave32 = one chunk of 16 consecutive points (same (b,n)).
// Geometry for the 16 points via one V_WMMA_F32_16X16X4_F32:
//   A (16x4): rows 0..2 = [comb_row | trans], rows 3..15 = 0
//   B (4x16): column j  = [xg*d, yg*d, d, 1] for point j
//   D (16x16): row i, col j = geom component i of point j -> lane j, VGPR i
__global__ __launch_bounds__(256) void bev_splat_kernel(
    const float* __restrict__ x,
    const float* __restrict__ intr,
    const float* __restrict__ extr,
    float* __restrict__ acc,
    int ws_layout)
{
    const int wave = (int)((blockIdx.x * blockDim.x + threadIdx.x) >> 5);
    const int lane = threadIdx.x & 31;
    const int l15  = lane & 15;
    const bool lo  = lane < 16;

    // chunk-uniform values -> force into SGPRs for scalar addressing
    const int P  = __builtin_amdgcn_readfirstlane(wave * 16); // first point
    const int bn = __builtin_amdgcn_readfirstlane(P / PER_BN);
    const int b  = __builtin_amdgcn_readfirstlane(bn / NN);

    // prefetch this chunk's 4KB feature block early (16 pts * 256B)
    const float* xbase = x + (size_t)P * CC;
    __builtin_prefetch((const char*)xbase + (size_t)lane * 128, 0, 1);

    // ---- K (3x3) and E (4x4) for this (b,n); 18 distinct, L2-resident ----
    const float* K = intr + (size_t)bn * 9;
    const float* E = extr + (size_t)bn * 16;
    const float k00=K[0],k01=K[1],k02=K[2],
                k10=K[3],k11=K[4],k12=K[5],
                k20=K[6],k21=K[7],k22=K[8];
    // adjugate inverse of K
    const float det = k00*(k11*k22 - k12*k21)
                    - k01*(k10*k22 - k12*k20)
                    + k02*(k10*k21 - k11*k20);
    const float id = 1.0f / det;
    const float i00 =  (k11*k22 - k12*k21)*id;
    const float i01 = -(k01*k22 - k02*k21)*id;
    const float i02 =  (k01*k12 - k02*k11)*id;
    const float i10 = -(k10*k22 - k12*k20)*id;
    const float i11 =  (k00*k22 - k02*k20)*id;
    const float i12 = -(k00*k12 - k02*k10)*id;
    const float i20 =  (k10*k21 - k11*k20)*id;
    const float i21 = -(k00*k21 - k01*k20)*id;
    const float i22 =  (k00*k11 - k01*k10)*id;

    const float r00=E[0], r01=E[1], r02=E[2],  t0=E[3];
    const float r10=E[4], r11=E[5], r12=E[6],  t1=E[7];
    const float r20=E[8], r21=E[9], r22=E[10], t2=E[11];

    // comb = R * inv(K)
    const float m00 = r00*i00 + r01*i10 + r02*i20;
    const float m01 = r00*i01 + r01*i11 + r02*i21;
    const float m02 = r00*i02 + r01*i12 + r02*i22;
    const float m10 = r10*i00 + r11*i10 + r12*i20;
    const float m11 = r10*i01 + r11*i11 + r12*i21;
    const float m12 = r10*i02 + r11*i12 + r12*i22;
    const float m20 = r20*i00 + r21*i10 + r22*i20;
    const float m21 = r20*i01 + r21*i11 + r22*i21;
    const float m22 = r20*i02 + r21*i12 + r22*i22;

    // ---- A operand (16x4 f32): lane l<16 -> {A[l][0], A[l][1]},
    //      lane l>=16 -> {A[l-16][2], A[l-16][3]} ----
    const float col0 = pick3(l15, m00, m10, m20);
    const float col1 = pick3(l15, m01, m11, m21);
    const float col2 = pick3(l15, m02, m12, m22);
    const float col3 = pick3(l15, t0,  t1,  t2 );
    v2f A; A.x = lo ? col0 : col2; A.y = lo ? col1 : col3;

    // ---- B operand (4x16 f32): VGPR0 = rows K=0/2, VGPR1 = rows K=1/3 ----
    const int rel = (P % PER_BN) + l15;
    const int dz  = rel / (DH * DW);
    const int rr  = rel - dz * (DH * DW);
    const int hh  = rr / DW;
    const int ww  = rr - hh * DW;
    const float dep = 2.0f + (float)dz;                  // arange(2,50,1)
    const float xg  = (float)ww * (479.0f / 59.0f);      // linspace(0,479,60)
    const float yg  = (float)hh * (223.0f / 27.0f);      // linspace(0,223,28)
    v2f Bm; Bm.x = lo ? xg * dep : dep; Bm.y = lo ? yg * dep : 1.0f;

    v8f Cz = {};
    // EXEC is all-ones here (exact grid, no divergence before this point)
    v8f Dv = __builtin_amdgcn_wmma_f32_16x16x4_f32(
        false, A, false, Bm, (short)0, Cz, false, false);

    // lane j (0..15) holds geom(point j): components in VGPRs 0,1,2
    const float gx = Dv[0], gy = Dv[1], gz = Dv[2];

    // voxelize — replicate astype(int32) truncation-toward-zero exactly
    const int vx = (int)((gx + 50.0f) / 0.5f);
    const int vy = (int)((gy + 50.0f) / 0.5f);
    const int vz = (int)((gz + 10.0f) / 20.0f);
    const bool msk = (vx >= 0 && vx < XD && vy >= 0 && vy < YD && vz == 0);

    // mask bits into an SGPR once; per-point skip becomes a scalar branch
    const unsigned long long bal = __ballot(msk);

    // ---- scatter: per point, all 32 lanes stream 2 channels each ----
    for (int j = 0; j < 16; ++j) {
        if (!((bal >> j) & 1ull)) continue;          // s_cbranch, no exec dance
        // broadcast voxel coords to SGPRs (j is uniform)
        const int vxj = __builtin_amdgcn_readlane(vx, j);
        const int vyj = __builtin_amdgcn_readlane(vy, j);
        const float2 f = reinterpret_cast<const float2*>(xbase + j * CC)[lane];
        if (ws_layout) {
            // channel-contiguous accumulator (B,X,Y,C): 2 cachelines/point
            float* dst = acc + ((((size_t)b * XD + vxj) * YD + vyj) * CC) + lane * 2;
            __hip_atomic_fetch_add(dst,     f.x, __ATOMIC_RELAXED, __HIP_MEMORY_SCOPE_AGENT);
            __hip_atomic_fetch_add(dst + 1, f.y, __ATOMIC_RELAXED, __HIP_MEMORY_SCOPE_AGENT);
        } else {
            // direct output layout (B,C,X,Y)
            const size_t o = ((((size_t)b * CC + lane * 2) * XD + vxj) * YD) + vyj;
            __hip_atomic_fetch_add(acc + o,                   f.x, __ATOMIC_RELAXED, __HIP_MEMORY_SCOPE_AGENT);
            __hip_atomic_fetch_add(acc + o + (size_t)XD * YD, f.y, __ATOMIC_RELAXED, __HIP_MEMORY_SCOPE_AGENT);
        }
    }
}

// (B,X,Y,C) accumulator -> (B,C,X,Y) output; coalesced writes, L2-covered reads
__global__ __launch_bounds__(256) void bev_transpose_kernel(
    const float* __restrict__ acc, float* __restrict__ out, size_t total)
{
    const size_t i = (size_t)blockIdx.x * blockDim.x + threadIdx.x;
    if (i >= total) return;
    const int yy = (int)(i % YD);
    size_t r = i / YD;
    const int xx = (int)(r % XD);  r /= XD;
    const int c  = (int)(r % CC);
    const int b  = (int)(r / CC);
    out[i] = acc[(((size_t)b * XD + xx) * YD + yy) * CC + c];
}

extern "C" void kernel_launch(void* const* d_in, const int* in_sizes, int n_in,
                              void* d_out, int out_size, void* d_ws, size_t ws_size,
                              hipStream_t stream) {
    const float* x    = (const float*)d_in[0];
    const float* intr = (const float*)d_in[1];
    const float* extr = (const float*)d_in[2];
    float* out = (float*)d_out;

    const size_t acc_bytes = ACC_ELEMS * sizeof(float);
    const bool use_ws = (ws_size >= acc_bytes);
    float* acc = use_ws ? (float*)d_ws : out;

    // zero the accumulator (harness poisons buffers with 0xAA)
    hipMemsetAsync(acc, 0, use_ws ? acc_bytes : (size_t)out_size * sizeof(float), stream);

    // 90,720 chunks, 8 waves/block -> 11,340 blocks (exact; EXEC stays full)
    bev_splat_kernel<<<NCHUNK / 8, 256, 0, stream>>>(x, intr, extr, acc, use_ws ? 1 : 0);

    if (use_ws) {
        const size_t total = ACC_ELEMS;
        bev_transpose_kernel<<<(int)((total + 255) / 256), 256, 0, stream>>>(acc, out, total);
    }
}